// BootstrapedMSEloss_71339406787253
// MI455X (gfx1250) — compile-verified
//
#include <hip/hip_runtime.h>
#include <stdint.h>

// Problem constants (match the JAX reference)
#define B_        64
#define CH        3
#define HW        65536
#define TOPK      200

// Kernel config
#define NTHREADS  1024
#define NWAVES    (NTHREADS / 32)          // 32 wave32 waves
#define NBINS     256                       // exponent-byte radix (values >= 0)
#define LCAP      2048                      // tie-bucket capacity
#define CHUNKS    (HW / 4 / NTHREADS)       // 16 float4 chunks per thread

// LDS layout (bytes) — total 303,296 B, fits CDNA5's 320 KB/WG
#define OFF_DIFF  0                                   // float[65536]      = 262144 B
#define OFF_HIST  262144                              // u32[NWAVES][256]  =  32768 B
#define OFF_LIST  (OFF_HIST + NWAVES * NBINS * 4)     // float[LCAP]       =   8192 B
#define OFF_RED   (OFF_LIST + LCAP * 4)               // float[32]         =    128 B
#define OFF_CTRL  (OFF_RED + 32 * 4)                  // u32[16]           =     64 B
#define LDS_BYTES (OFF_CTRL + 64)

static_assert(CHUNKS == 16, "chunk math");
static_assert(LDS_BYTES <= 320 * 1024, "LDS budget");

typedef __attribute__((ext_vector_type(4))) float f32x4;

// Phase-1 step K: consume the async-staged t2 chunk from the diff region,
// combine with 5 register-streamed planes, write diff back in place.
// The asm block is self-contained: waits for this lane's K-th async DMA
// (async loads complete in order per wave), loads it, and fully retires the
// hidden ds_load so the compiler's DS-counter bookkeeping stays exact.
template<int K>
__device__ __forceinline__ void phase1_steps(const float4* __restrict__ p0,
                                             const float4* __restrict__ p1,
                                             const float4* __restrict__ p2,
                                             const float4* __restrict__ t0,
                                             const float4* __restrict__ t1,
                                             float4* diff4, unsigned diffbase,
                                             unsigned* myhist, int tid)
{
    if constexpr (K < CHUNKS) {
        const int j = K * NTHREADS + tid;
        const float4 a0 = p0[j], a1 = p1[j], a2 = p2[j];
        const float4 c0 = t0[j], c1 = t1[j];

        f32x4 c2;
        unsigned laddr = diffbase + (unsigned)j * 16u;
        asm volatile("s_wait_asynccnt %1\n\t"
                     "ds_load_b128 %0, %2\n\t"
                     "s_wait_dscnt 0x0"
                     : "=v"(c2)
                     : "i"(CHUNKS - 1 - K), "v"(laddr));

        float4 d;
        { float e0 = c0.x - a0.x, e1 = c1.x - a1.x, e2 = c2.x - a2.x;
          d.x = fmaf(e0, e0, fmaf(e1, e1, e2 * e2)); }
        { float e0 = c0.y - a0.y, e1 = c1.y - a1.y, e2 = c2.y - a2.y;
          d.y = fmaf(e0, e0, fmaf(e1, e1, e2 * e2)); }
        { float e0 = c0.z - a0.z, e1 = c1.z - a1.z, e2 = c2.z - a2.z;
          d.z = fmaf(e0, e0, fmaf(e1, e1, e2 * e2)); }
        { float e0 = c0.w - a0.w, e1 = c1.w - a1.w, e2 = c2.w - a2.w;
          d.w = fmaf(e0, e0, fmaf(e1, e1, e2 * e2)); }

        diff4[j] = d;                                   // same-wave DS: in order
        // values >= 0 -> exponent byte is a monotone radix key
        atomicAdd(&myhist[__float_as_uint(d.x) >> 23], 1u);
        atomicAdd(&myhist[__float_as_uint(d.y) >> 23], 1u);
        atomicAdd(&myhist[__float_as_uint(d.z) >> 23], 1u);
        atomicAdd(&myhist[__float_as_uint(d.w) >> 23], 1u);

        phase1_steps<K + 1>(p0, p1, p2, t0, t1, diff4, diffbase, myhist, tid);
    }
}

// One workgroup per batch row: stream 6 channel-planes from HBM once
// (5 via VGPR-return loads, 1 via async DMA straight into the LDS diff
// region), then radix-select + exact tie resolution entirely in LDS.
__global__ __launch_bounds__(NTHREADS)
void bmse_row_topk_kernel(const float* __restrict__ pred,
                          const float* __restrict__ tgt,
                          float* __restrict__ row_sums)
{
    extern __shared__ unsigned char smem[];
    float*    diff  = (float*)(smem + OFF_DIFF);
    unsigned* hist  = (unsigned*)(smem + OFF_HIST);
    float*    list  = (float*)(smem + OFF_LIST);
    float*    red   = (float*)(smem + OFF_RED);
    unsigned* ctrl  = (unsigned*)(smem + OFF_CTRL);
    float*    ctrlf = (float*)(smem + OFF_CTRL);
    // ctrl[0]=threshold bin T, ctrl[1]=k_rem, ctrl[2]=tie count, ctrlf[3]=above-sum

    const int tid  = threadIdx.x;
    const int wave = tid >> 5;
    const int b    = blockIdx.x;

    const size_t base = (size_t)b * (CH * HW);
    const float4* p0 = (const float4*)(pred + base);
    const float4* p1 = (const float4*)(pred + base + HW);
    const float4* p2 = (const float4*)(pred + base + (size_t)2 * HW);
    const float4* t0 = (const float4*)(tgt  + base);
    const float4* t1 = (const float4*)(tgt  + base + HW);
    const float4* t2 = (const float4*)(tgt  + base + (size_t)2 * HW);

    float4*  diff4   = (float4*)diff;
    // low 32 bits of a generic LDS pointer == DS byte address
    const unsigned diffbase = (unsigned)(uintptr_t)(const void*)diff;

    // ---- kick off async DMA of the t2 plane into the (currently dead)
    //      diff region: 16 x global_load_async_to_lds_b128 per lane ----
#pragma unroll
    for (int k = 0; k < CHUNKS; ++k) {
        const int j = k * NTHREADS + tid;
        const float4* g = t2 + j;
        unsigned l = diffbase + (unsigned)j * 16u;
        asm volatile("global_load_async_to_lds_b128 %0, %1, off"
                     :: "v"(l), "v"(g)
                     : "memory");
    }

    // ---- init histograms / control (overlaps with DMA) ----
#pragma unroll
    for (int i = 0; i < (NWAVES * NBINS) / NTHREADS; ++i)
        hist[i * NTHREADS + tid] = 0u;
    if (tid < 16) ctrl[tid] = 0u;
    __syncthreads();

    // ---- phase 1: stream inputs, diff -> LDS (in place), per-wave histogram ----
    phase1_steps<0>(p0, p1, p2, t0, t1, diff4, diffbase, hist + wave * NBINS, tid);
    __syncthreads();

    // ---- phase 2: merge per-wave histograms (deterministic) ----
    if (tid < NBINS) {
        unsigned s = 0;
#pragma unroll
        for (int w = 0; w < NWAVES; ++w) s += hist[w * NBINS + tid];
        hist[tid] = s;
    }
    __syncthreads();

    // ---- phase 3: find threshold bucket ----
    if (tid == 0) {
        unsigned cum = 0, T = 0, above = 0;
        for (int bin = NBINS - 1; bin >= 0; --bin) {
            const unsigned c = hist[bin];
            if (cum + c >= TOPK) { T = (unsigned)bin; above = cum; break; }
            cum += c;
        }
        ctrl[0] = T;
        ctrl[1] = TOPK - above;   // 1..TOPK values still needed from bucket T
    }
    __syncthreads();

    const unsigned T    = ctrl[0];
    const unsigned krem = ctrl[1];

    // ---- phase 4: sum strictly-above values; collect tie-bucket values ----
    float lsum = 0.f;
#pragma unroll 4
    for (int k = 0; k < CHUNKS; ++k) {
        const int j = k * NTHREADS + tid;
        const float4 d = diff4[j];                      // ds_load_b128
        const float vv[4] = {d.x, d.y, d.z, d.w};
#pragma unroll
        for (int q = 0; q < 4; ++q) {
            const float v = vv[q];
            const unsigned bin = __float_as_uint(v) >> 23;
            if (bin > T) {
                lsum += v;
            } else if (bin == T) {
                const unsigned pos = atomicAdd(&ctrl[2], 1u);
                if (pos < LCAP) list[pos] = v;
            }
        }
    }
    // deterministic block reduction of the above-threshold sum
#pragma unroll
    for (int off = 16; off > 0; off >>= 1) lsum += __shfl_xor(lsum, off, 32);
    if ((tid & 31) == 0) red[wave] = lsum;
    __syncthreads();

    if (wave == 0) {
        float s = red[tid];
#pragma unroll
        for (int off = 16; off > 0; off >>= 1) s += __shfl_xor(s, off, 32);
        if (tid == 0) ctrlf[3] = s;
    }
    const unsigned L = (ctrl[2] < (unsigned)LCAP) ? ctrl[2] : (unsigned)LCAP;
    for (unsigned i = tid; i < LCAP; i += NTHREADS)
        if (i >= L) list[i] = -1.0f;                    // pad below any real value

    // ---- phase 5: bitonic sort tie list descending (66 stages) ----
    for (unsigned kk = 2; kk <= LCAP; kk <<= 1) {
        for (unsigned j = kk >> 1; j > 0; j >>= 1) {
            __syncthreads();
            for (unsigned i = tid; i < LCAP; i += NTHREADS) {
                const unsigned ixj = i ^ j;
                if (ixj > i) {
                    const float x = list[i], y = list[ixj];
                    const bool desc = ((i & kk) == 0);
                    if (desc ? (x < y) : (x > y)) { list[i] = y; list[ixj] = x; }
                }
            }
        }
    }
    __syncthreads();

    // ---- phase 6: add the top k_rem tie values, emit per-row sum ----
    if (tid == 0) {
        float s = ctrlf[3];
        for (unsigned i = 0; i < krem; ++i) s += list[i];
        row_sums[b] = s;
    }
}

__global__ void bmse_finalize_kernel(const float* __restrict__ row_sums,
                                     float* __restrict__ out)
{
    if (blockIdx.x == 0 && threadIdx.x == 0) {
        float s = 0.f;
        for (int i = 0; i < B_; ++i) s += row_sums[i];
        out[0] = s * (1.0f / (float)(B_ * TOPK));
    }
}

extern "C" void kernel_launch(void* const* d_in, const int* in_sizes, int n_in,
                              void* d_out, int out_size, void* d_ws, size_t ws_size,
                              hipStream_t stream)
{
    (void)in_sizes; (void)n_in; (void)out_size; (void)ws_size;
    const float* pred = (const float*)d_in[0];
    const float* tgt  = (const float*)d_in[1];
    float* row_sums   = (float*)d_ws;   // 64 floats of scratch

    bmse_row_topk_kernel<<<B_, NTHREADS, LDS_BYTES, stream>>>(pred, tgt, row_sums);
    bmse_finalize_kernel<<<1, 32, 0, stream>>>(row_sums, (float*)d_out);
}